// LSS_Core_Complete_30107720745184
// MI455X (gfx1250) — compile-verified
//
#include <hip/hip_runtime.h>
#include <hip/hip_bf16.h>

// ---------------- problem constants (match reference) ----------------
#define B_    2
#define N_    6
#define CIN   128
#define H_    16
#define W_    44
#define D_    41
#define C_    64
#define NX0   200
#define NX1   200
#define HW_   (H_*W_)            // 704
#define PTOT  (B_*N_*HW_)        // 8448 pixels
#define NOUT  (D_+C_)            // 105
#define BN_EPS 1e-5f

typedef __attribute__((ext_vector_type(2))) float v2f;
typedef __attribute__((ext_vector_type(8))) float v8f;

// fp32 WMMA: D(16x16,f32) = A(16x4,f32) * B(4x16,f32) + C
#define WMMA_F32(a, b, c) \
  __builtin_amdgcn_wmma_f32_16x16x4_f32(false, (a), false, (b), (short)0, (c), false, false)

// ---------------- small utility kernels ----------------
__global__ void zero_f32(float* __restrict__ p, int n) {
  int i = blockIdx.x * 256 + threadIdx.x;
  if (i < n) p[i] = 0.f;
}

// w_bev1 (o, i, 3, 3) -> wt (tap=r*3+s, i, o)  : makes WMMA B-fragment loads lane-contiguous
__global__ void transpose_wbev1(const float* __restrict__ w, float* __restrict__ wt) {
  int e = blockIdx.x * 256 + threadIdx.x;
  if (e >= C_ * C_ * 9) return;
  int s = e % 3; int t = e / 3;
  int r = t % 3; t /= 3;
  int i = t % C_;
  int o = t / C_;
  wt[((r * 3 + s) * C_ + i) * C_ + o] = w[e];
}

// ---------------- stage 1+2: cam GEMM (WMMA) + softmax + voxel scatter ----------------
// one wave (32 threads) per 16-pixel tile; GEMM M=16, N=112 (105 padded), K=128
__global__ __launch_bounds__(32) void cam_gemm_scatter(
    const float* __restrict__ x, const float* __restrict__ coords,
    const float* __restrict__ w_cam, const float* __restrict__ b_cam,
    float* __restrict__ bev)
{
  __shared__ float xs[CIN * 16];   // [c][pix]  channel-major input tile (8 KB)
  __shared__ float fs[16 * 112];   // [pix][o]  feat -> probs/context        (7 KB)

  const int lane = threadIdx.x;
  const int lm   = lane & 15;
  const int half = lane >> 4;

  const int p0  = blockIdx.x * 16;      // tiles never cross bn (HW_ % 16 == 0)
  const int bn  = p0 / HW_;
  const int hw0 = p0 - bn * HW_;
  const int b   = bn / N_;

  __builtin_prefetch(&w_cam[lane * CIN], 0, 3);   // global_prefetch_b8 warmup of weights

  // cooperative stage of the 16x128 input tile (coalesced: 16 contiguous hw per channel)
  const float* xp = x + (size_t)bn * CIN * HW_ + hw0;
  #pragma unroll 4
  for (int it = 0; it < 64; ++it) {
    int flat = it * 32 + lane;
    int c = flat >> 4, pix = flat & 15;
    xs[c * 16 + pix] = xp[c * HW_ + pix];
  }
  __syncthreads();

  v8f acc[7];
  #pragma unroll
  for (int nt = 0; nt < 7; ++nt) acc[nt] = (v8f){0,0,0,0,0,0,0,0};

  for (int k0 = 0; k0 < CIN; k0 += 4) {
    const int ca = k0 + half * 2;                 // K pair owned by this half-wave
    v2f a;
    a.x = xs[ca * 16 + lm];
    a.y = xs[(ca + 1) * 16 + lm];
    #pragma unroll
    for (int nt = 0; nt < 7; ++nt) {
      int o = nt * 16 + lm;
      int ow = (o < NOUT) ? o : (NOUT - 1);
      float msk = (o < NOUT) ? 1.f : 0.f;         // pad columns contribute zero
      v2f bb;
      bb.x = w_cam[ow * CIN + ca]     * msk;
      bb.y = w_cam[ow * CIN + ca + 1] * msk;
      acc[nt] = WMMA_F32(a, bb, acc[nt]);
    }
  }

  // spill feat(+bias) to LDS in [pix][o] layout
  #pragma unroll
  for (int nt = 0; nt < 7; ++nt) {
    int o = nt * 16 + lm;
    if (o < NOUT) {
      float bias = b_cam[o];
      #pragma unroll
      for (int v = 0; v < 8; ++v) {
        int pix = v + half * 8;
        fs[pix * 112 + o] = acc[nt][v] + bias;
      }
    }
  }
  __syncthreads();

  // softmax over the 41 depth bins, one pixel per lane (lanes 0..15)
  if (lane < 16) {
    float m = -3.4e38f;
    for (int d = 0; d < D_; ++d) m = fmaxf(m, fs[lane * 112 + d]);
    float s = 0.f;
    for (int d = 0; d < D_; ++d) {
      float e = __expf(fs[lane * 112 + d] - m);
      fs[lane * 112 + d] = e;
      s += e;
    }
    float inv = 1.f / s;
    for (int d = 0; d < D_; ++d) fs[lane * 112 + d] *= inv;
  }
  __syncthreads();

  // scatter: point (pix,d) -> bev[b, c, iy, ix] += prob * context[c]; 2 channels/lane
  for (int pix = 0; pix < 16; ++pix) {
    const int hw = hw0 + pix;
    const float* cp = coords + ((size_t)bn * D_ * HW_ + hw) * 3;
    const float c0 = fs[pix * 112 + D_ + lane];
    const float c1 = fs[pix * 112 + D_ + lane + 32];
    for (int d = 0; d < D_; ++d) {
      float cx = cp[(size_t)d * HW_ * 3 + 0];
      float cy = cp[(size_t)d * HW_ * 3 + 1];
      float cz = cp[(size_t)d * HW_ * 3 + 2];
      int ix = (int)((cx + 50.f) * 2.f);          // /0.5 == *2 (exact)
      int iy = (int)((cy + 50.f) * 2.f);
      int iz = (int)((cz + 10.f) / 20.f);
      if (ix >= 0 && ix < NX0 && iy >= 0 && iy < NX1 && iz == 0) {
        float pr = fs[pix * 112 + d];
        size_t base = ((size_t)(b * C_) * NX1 + iy) * NX0 + ix;
        atomicAdd(&bev[base + (size_t)lane        * NX1 * NX0], pr * c0);
        atomicAdd(&bev[base + (size_t)(lane + 32) * NX1 * NX0], pr * c1);
      }
    }
  }
}

// ---------------- stage 3: 3x3 conv as implicit GEMM (WMMA) + BN stats ----------------
// one wave per (b, y, 16-wide x tile); K = 9 taps x 64 cin; N = 64 cout (4 tiles)
#define XTILES 13   // ceil(200/16)
__global__ __launch_bounds__(32) void bev_conv_wmma(
    const float* __restrict__ bev, const float* __restrict__ wt,
    float* __restrict__ h, float* __restrict__ sums /* [0..63] sum, [64..127] sumsq */)
{
  __shared__ float ts[C_ * 3 * 18];   // [cin][row 3][col 18] halo tile (13.8 KB)

  const int lane = threadIdx.x;
  const int lm   = lane & 15;
  const int half = lane >> 4;

  int blk = blockIdx.x;
  const int xt = blk % XTILES; blk /= XTILES;
  const int y  = blk % NX1;    blk /= NX1;
  const int b  = blk;
  const int x0 = xt * 16;

  // stage halo: 64ch x 3 rows x 18 cols, zero padded at borders (3456 elems / 32 lanes)
  #pragma unroll 4
  for (int it = 0; it < 108; ++it) {
    int e = it * 32 + lane;
    int c = e / 54; int rem = e - c * 54;
    int r = rem / 18; int col = rem - r * 18;
    int gy = y + r - 1, gx = x0 + col - 1;
    float v = 0.f;
    if (gy >= 0 && gy < NX1 && gx >= 0 && gx < NX0)
      v = bev[((size_t)(b * C_ + c) * NX1 + gy) * NX0 + gx];
    ts[(c * 3 + r) * 18 + col] = v;
  }
  __syncthreads();

  v8f acc[4];
  #pragma unroll
  for (int nt = 0; nt < 4; ++nt) acc[nt] = (v8f){0,0,0,0,0,0,0,0};

  for (int r = 0; r < 3; ++r) {
    for (int s = 0; s < 3; ++s) {
      const float* wtap = wt + (size_t)(r * 3 + s) * C_ * C_;   // [cin][cout]
      for (int k0 = 0; k0 < C_; k0 += 4) {
        const int ca = k0 + half * 2;
        v2f a;
        a.x = ts[(ca * 3 + r) * 18 + lm + s];
        a.y = ts[((ca + 1) * 3 + r) * 18 + lm + s];
        #pragma unroll
        for (int nt = 0; nt < 4; ++nt) {
          v2f bb;
          bb.x = wtap[ca * C_ + nt * 16 + lm];
          bb.y = wtap[(ca + 1) * C_ + nt * 16 + lm];
          acc[nt] = WMMA_F32(a, bb, acc[nt]);
        }
      }
    }
  }

  // epilogue: write h, accumulate per-channel sum / sumsq for BN
  #pragma unroll
  for (int nt = 0; nt < 4; ++nt) {
    const int o = nt * 16 + lm;
    float s1 = 0.f, s2 = 0.f;
    #pragma unroll
    for (int v = 0; v < 8; ++v) {
      int gx = x0 + v + half * 8;
      float val = acc[nt][v];
      if (gx < NX0) {
        h[((size_t)(b * C_ + o) * NX1 + y) * NX0 + gx] = val;
        s1 += val;
        s2 += val * val;
      }
    }
    atomicAdd(&sums[o],      s1);
    atomicAdd(&sums[C_ + o], s2);
  }
}

// ---------------- stage 4: finalize BN to per-channel scale/shift ----------------
__global__ void bn_finalize(const float* __restrict__ sums,
                            const float* __restrict__ gamma,
                            const float* __restrict__ beta,
                            float* __restrict__ sc /* [0..63] scale, [64..127] shift */)
{
  int c = threadIdx.x;
  if (c < C_) {
    const float inv_cnt = 1.f / (float)(B_ * NX0 * NX1);
    float mean = sums[c] * inv_cnt;
    float var  = sums[C_ + c] * inv_cnt - mean * mean;
    float scal = gamma[c] * rsqrtf(var + BN_EPS);
    sc[c]      = scal;
    sc[C_ + c] = beta[c] - mean * scal;
  }
}

// ---------------- stage 5: BN + ReLU + 1x64 head ----------------
__global__ __launch_bounds__(256) void bev_head(
    const float* __restrict__ hbuf, const float* __restrict__ sc,
    const float* __restrict__ w2, const float* __restrict__ b2,
    float* __restrict__ out)
{
  int pid = blockIdx.x * 256 + threadIdx.x;
  if (pid >= B_ * NX0 * NX1) return;
  int b  = pid / (NX0 * NX1);
  int yx = pid - b * (NX0 * NX1);
  float accv = b2[0];
  #pragma unroll 8
  for (int c = 0; c < C_; ++c) {
    float hv = hbuf[(size_t)(b * C_ + c) * (NX0 * NX1) + yx];
    hv = fmaxf(hv * sc[c] + sc[C_ + c], 0.f);
    accv += hv * w2[c];
  }
  out[pid] = accv;
}

// ---------------- launch ----------------
extern "C" void kernel_launch(void* const* d_in, const int* in_sizes, int n_in,
                              void* d_out, int out_size, void* d_ws, size_t ws_size,
                              hipStream_t stream) {
  (void)in_sizes; (void)n_in; (void)out_size; (void)ws_size;

  const float* x      = (const float*)d_in[0];
  const float* coords = (const float*)d_in[1];
  const float* w_cam  = (const float*)d_in[2];
  const float* b_cam  = (const float*)d_in[3];
  const float* w_bev1 = (const float*)d_in[4];
  const float* gamma  = (const float*)d_in[5];
  const float* beta   = (const float*)d_in[6];
  const float* w_bev2 = (const float*)d_in[7];
  const float* b_bev2 = (const float*)d_in[8];
  float* out = (float*)d_out;

  // workspace layout (floats)
  const size_t BEV_N = (size_t)B_ * C_ * NX1 * NX0;   // 5,120,000
  float* ws   = (float*)d_ws;
  float* bev  = ws;                                   // scatter target
  float* hbuf = ws + BEV_N;                           // conv output
  float* wt   = ws + 2 * BEV_N;                       // transposed conv weights (36,864)
  float* sums = wt + (size_t)C_ * C_ * 9;             // 128
  float* sc   = sums + 2 * C_;                        // 128

  // 0) zero scatter grid + stats
  zero_f32<<<(int)((BEV_N + 255) / 256), 256, 0, stream>>>(bev, (int)BEV_N);
  zero_f32<<<1, 256, 0, stream>>>(sums, 2 * C_);

  // 1) weight transpose for implicit-GEMM conv
  transpose_wbev1<<<(C_ * C_ * 9 + 255) / 256, 256, 0, stream>>>(w_bev1, wt);

  // 2) cam GEMM + softmax + scatter  (528 one-wave tiles)
  cam_gemm_scatter<<<PTOT / 16, 32, 0, stream>>>(x, coords, w_cam, b_cam, bev);

  // 3) 3x3 conv via WMMA + BN statistics (2 * 200 * 13 tiles)
  bev_conv_wmma<<<B_ * NX1 * XTILES, 32, 0, stream>>>(bev, wt, hbuf, sums);

  // 4) finalize BN
  bn_finalize<<<1, 64, 0, stream>>>(sums, gamma, beta, sc);

  // 5) BN + ReLU + head projection
  bev_head<<<(B_ * NX0 * NX1 + 255) / 256, 256, 0, stream>>>(hbuf, sc, w_bev2, b_bev2, out);
}